// Layer_Attention_MultiHead_GAT_16535624089991
// MI455X (gfx1250) — compile-verified
//
#include <hip/hip_runtime.h>
#include <hip/hip_bf16.h>

// GAT multi-head attention layer for MI455X (gfx1250, wave32, WMMA).
#define GN 4096   // nodes
#define GF 512    // input features
#define GFP 64    // per-head output features
#define GK 8      // heads
#define ALPHA 0.2f

typedef __attribute__((ext_vector_type(2))) float v2f;
typedef __attribute__((ext_vector_type(8))) float v8f;

// ---------------------------------------------------------------------------
// Kernel 1: h[k][n][p] = sum_f node[n][f] * W[k][f][p]
// One wave computes a 16x16 tile with V_WMMA_F32_16X16X4_F32 (K-loop step 4).
// Result is stored in PAIRED-ROW layout hp[k][n/2][p][2] so that the
// attention kernel's WMMA B operands (row pair n,n+1 at one column) become
// single coalesced b64 loads.
//
// f32 WMMA lane layouts (ISA 7.12.2):
//   A (16x4):  VGPR0: lanes0-15 K=0, lanes16-31 K=2 ; VGPR1: K=1/K=3 ; M=lane%16
//   B (4x16):  VGPR0: lanes0-15 row K=0, lanes16-31 row K=2 ; N=lane%16
//   C/D:       VGPR v: lanes0-15 M=v, lanes16-31 M=v+8 ; N=lane%16
// ---------------------------------------------------------------------------
__global__ void __launch_bounds__(32)
gat_h_gemm(const float* __restrict__ node, const float* __restrict__ W,
           float* __restrict__ hp)
{
    const int lane = threadIdx.x;
    const int half = lane >> 4;
    const int l16  = lane & 15;
    const int p0 = blockIdx.x * 16;
    const int m0 = blockIdx.y * 16;
    const int k  = blockIdx.z;

    const float* __restrict__ Arow = node + (size_t)(m0 + l16) * GF;
    const float* __restrict__ Wk   = W + (size_t)k * GF * GFP;

    v8f acc = {};
#pragma unroll 4
    for (int f0 = 0; f0 < GF; f0 += 4) {
        const int fa = f0 + 2 * half;
        v2f a, b;
        a[0] = Arow[fa];
        a[1] = Arow[fa + 1];
        b[0] = Wk[(size_t)fa       * GFP + p0 + l16];
        b[1] = Wk[(size_t)(fa + 1) * GFP + p0 + l16];
        acc = __builtin_amdgcn_wmma_f32_16x16x4_f32(
            false, a, false, b, (short)0, acc, false, false);
    }

    // paired-row store: rows (m, m+1) interleaved in the last dimension
    float* __restrict__ hpk = hp + (size_t)k * GN * GFP;
#pragma unroll
    for (int u = 0; u < 4; ++u) {
        v2f w;
        w[0] = acc[2 * u];
        w[1] = acc[2 * u + 1];
        const size_t rowpair = (size_t)(m0 >> 1) + 4 * half + u;
        *(v2f*)(hpk + (rowpair * GFP + p0 + l16) * 2) = w;
    }
}

// ---------------------------------------------------------------------------
// Kernel 2: f1[k][n] = dot(h[k][n][:], a_src[k]) ; f2 likewise with a_dst.
// Reads the paired-row layout.
// ---------------------------------------------------------------------------
__global__ void gat_f12(const float* __restrict__ hp, const float* __restrict__ a,
                        float* __restrict__ f1, float* __restrict__ f2)
{
    const int tid = blockIdx.x * blockDim.x + threadIdx.x;
    if (tid >= GK * GN) return;
    const int k = tid / GN;
    const int n = tid % GN;
    const float* __restrict__ hr =
        hp + (size_t)k * GN * GFP + (size_t)(n >> 1) * GFP * 2 + (n & 1);
    const float* __restrict__ as = a + (size_t)k * 2 * GFP;
    const float* __restrict__ ad = as + GFP;
    float s1 = 0.f, s2 = 0.f;
#pragma unroll 8
    for (int p = 0; p < GFP; ++p) {
        const float hv = hr[2 * p];
        s1 += hv * as[p];
        s2 += hv * ad[p];
    }
    f1[tid] = s1;
    f2[tid] = s2;
}

// ---------------------------------------------------------------------------
// Kernel 2b: gmax[k] = max_n f1[k][n]  (per-head softmax stability bound;
// leaky_relu is monotonic so max_n e[m,n] = leaky(f2[m] + gmax[k])).
// ---------------------------------------------------------------------------
__global__ void __launch_bounds__(256)
gat_gmax(const float* __restrict__ f1, float* __restrict__ gmax)
{
    __shared__ float red[256];
    const int k = blockIdx.x;
    const int t = threadIdx.x;
    const float* __restrict__ f1k = f1 + (size_t)k * GN;
    float m = -__builtin_inff();
    for (int n = t; n < GN; n += 256) m = fmaxf(m, f1k[n]);
    red[t] = m;
    __syncthreads();
#pragma unroll
    for (int s = 128; s > 0; s >>= 1) {
        if (t < s) red[t] = fmaxf(red[t], red[t + s]);
        __syncthreads();
    }
    if (t == 0) gmax[k] = red[0];
}

// ---------------------------------------------------------------------------
// Kernel 3: streaming softmax-attention, shuffle-free inner loop.
// One wave: (head k, 16 rows). Per 16-column tile:
//   p = exp(clamp(leaky(f2[m]+f1[n]) - EM[m], -80))   (masked -> exp(-80)~0)
//   per-lane partial row sums (reduced once at the end)
//   O(16x64) += P(16x16) @ h_tile(16x64) via 16x V_WMMA_F32_16X16X4_F32
// P goes through LDS (C-layout -> A-layout); B operands are coalesced b64
// loads from the paired-row h layout.
// ---------------------------------------------------------------------------
__global__ void __launch_bounds__(32)
gat_attn(const int* __restrict__ adj, const float* __restrict__ hp,
         const float* __restrict__ f1, const float* __restrict__ f2,
         const float* __restrict__ gmax, float* __restrict__ out)
{
    __shared__ alignas(16) float pS[16][20];  // padded vs bank conflicts

    const int lane = threadIdx.x;
    const int half = lane >> 4;
    const int l16  = lane & 15;
    const int m0 = blockIdx.x * 16;
    const int k  = blockIdx.y;

    const float* __restrict__ f1k = f1 + (size_t)k * GN;
    const float* __restrict__ f2k = f2 + (size_t)k * GN;
    const float* __restrict__ hpk = hp + (size_t)k * GN * GFP;
    const float gm = gmax[k];

    float EM[8], rsum[8];
#pragma unroll
    for (int v = 0; v < 8; ++v) {
        const float fm = f2k[m0 + v + 8 * half];
        const float x  = fm + gm;
        EM[v]   = fmaxf(x, ALPHA * x);   // row-wise softmax shift (>= row max)
        rsum[v] = 0.f;
    }
    float fmr[8];
#pragma unroll
    for (int v = 0; v < 8; ++v) fmr[v] = f2k[m0 + v + 8 * half];

    v8f O[4] = {v8f{}, v8f{}, v8f{}, v8f{}};

    for (int n0 = 0; n0 < GN; n0 += 16) {
        const float f1n = f1k[n0 + l16];

        // probabilities in C-layout: lane holds rows m=v+8*half, col n0+l16
        float p[8];
#pragma unroll
        for (int v = 0; v < 8; ++v) {
            const int av = adj[(size_t)(m0 + v + 8 * half) * GN + n0 + l16];
            float x = fmr[v] + f1n;
            x = fmaxf(x, ALPHA * x);                // leaky relu
            x = (av > 0) ? (x - EM[v]) : -80.f;     // mask + shift
            x = fmaxf(x, -80.f);                    // keep exp() finite & >0
            const float pe = __expf(x);
            p[v] = pe;
            rsum[v] += pe;                          // per-lane partial sum
        }

        // stage P through LDS: C-layout -> A-layout
#pragma unroll
        for (int v = 0; v < 8; ++v)
            pS[v + 8 * half][l16] = p[v];
        __syncthreads();

        v2f A[4];
#pragma unroll
        for (int c = 0; c < 4; ++c)
            A[c] = *(const v2f*)&pS[l16][c * 4 + 2 * half];
        __syncthreads();

        // O(16x64) += P @ h_tile : B = row pair (nr, nr+1) via one b64 load
#pragma unroll
        for (int t = 0; t < 4; ++t) {
#pragma unroll
            for (int c = 0; c < 4; ++c) {
                const size_t rowpair = (size_t)(n0 >> 1) + c * 2 + half;
                const v2f B = *(const v2f*)(hpk + (rowpair * GFP + t * 16 + l16) * 2);
                O[t] = __builtin_amdgcn_wmma_f32_16x16x4_f32(
                    false, A[c], false, B, (short)0, O[t], false, false);
            }
        }
    }

    // final row-sum reduction across the 16 lanes of each half
#pragma unroll
    for (int v = 0; v < 8; ++v) {
        float s = rsum[v];
#pragma unroll
        for (int msk = 1; msk < 16; msk <<= 1)
            s += __shfl_xor(s, msk, 16);
        rsum[v] = s;
    }

    // epilogue: divide by row sum, write out[n][k*FP + p]
#pragma unroll
    for (int t = 0; t < 4; ++t) {
#pragma unroll
        for (int v = 0; v < 8; ++v) {
            out[(size_t)(m0 + v + 8 * half) * (GK * GFP) + k * GFP + t * 16 + l16]
                = O[t][v] / rsum[v];
        }
    }
}

// ---------------------------------------------------------------------------
extern "C" void kernel_launch(void* const* d_in, const int* in_sizes, int n_in,
                              void* d_out, int out_size, void* d_ws, size_t ws_size,
                              hipStream_t stream)
{
    const float* node = (const float*)d_in[0];  // (N, F) f32
    const int*   adj  = (const int*)d_in[1];    // (N, N) i32
    const float* W    = (const float*)d_in[2];  // (K, F, FP) f32
    const float* a    = (const float*)d_in[3];  // (K, 2*FP, 1) f32
    float* out = (float*)d_out;                 // (N, K*FP) f32

    // ws layout: hp (K*N*FP f32, paired-row, 8 MB) | f1 | f2 | gmax
    float* hp   = (float*)d_ws;
    float* f1   = hp + (size_t)GK * GN * GFP;
    float* f2   = f1 + (size_t)GK * GN;
    float* gmax = f2 + (size_t)GK * GN;

    gat_h_gemm<<<dim3(GFP / 16, GN / 16, GK), 32, 0, stream>>>(node, W, hp);
    gat_f12<<<(GK * GN + 255) / 256, 256, 0, stream>>>(hp, a, f1, f2);
    gat_gmax<<<GK, 256, 0, stream>>>(f1, gmax);
    gat_attn<<<dim3(GN / 16, GK), 32, 0, stream>>>(adj, hp, f1, f2, gmax, out);
}